// Model_41266045780578
// MI455X (gfx1250) — compile-verified
//
#include <hip/hip_runtime.h>
#include <math.h>

#define IN_DIM 128
#define H_DIM 256
#define OUT_DIM 128
#define B_DIM 256
#define T_DIM 512
#define NCHUNK 256
#define EPS_BN 1e-5f

typedef __attribute__((ext_vector_type(2))) float v2f;
typedef __attribute__((ext_vector_type(8))) float v8f;

__device__ __forceinline__ v8f wmma_f32(v2f a, v2f b, v8f c) {
  // V_WMMA_F32_16X16X4_F32 : D = A(16x4) x B(4x16) + C(16x16), all fp32
  return __builtin_amdgcn_wmma_f32_16x16x4_f32(
      /*neg_a=*/false, a, /*neg_b=*/false, b,
      /*c_mod=*/(short)0, c, /*reuse_a=*/false, /*reuse_b=*/false);
}

__device__ __forceinline__ float sigmoid_f(float x) {
  return 1.0f / (1.0f + expf(-x));
}

// ---------------------------------------------------------------------------
// One GRU-cell timestep: computes Hnew = GRUCell(Ain_row(t), Hprev).
// Grid: (B/16, H/16) blocks of a single wave32. Each wave owns one 16x16
// output tile and 4 fp32 WMMA accumulators:
//   accR   = x@Wih_r.T + h@Whh_r.T      accZ  = x@Wih_z.T + h@Whh_z.T
//   accGin = x@Wih_n.T                  accGhn = h@Whh_n.T
// A-layout (fp32 16x4, ISA 7.12.2): lane<16 -> row=lane, {K,K+1};
//                                   lane>=16 -> row=lane-16, {K+2,K+3}.
// B-layout mirrored: lane<16 -> col=lane, {K,K+1}; lane>=16 -> {K+2,K+3}.
// C/D: c[v] = element(M = v + 8*(lane>>4), N = lane&15).
// ---------------------------------------------------------------------------
__global__ __launch_bounds__(32) void gru_step_kernel(
    const float* __restrict__ Ain, int lda, int Kx,
    const float* __restrict__ Hprev,
    const float* __restrict__ Wih, const float* __restrict__ Whh,
    const float* __restrict__ bih, const float* __restrict__ bhh,
    float* __restrict__ Hnew,
    float* __restrict__ seqOut, int seqStride) {
  const int lane = (int)threadIdx.x;
  const int lm   = lane & 15;
  const int half = lane >> 4;
  const int r0   = (int)blockIdx.x << 4;
  const int c0   = (int)blockIdx.y << 4;

  v8f accR = {}; v8f accZ = {}; v8f accGin = {}; v8f accGhn = {};

  // ---- input-gate GEMM: gi = x_t @ Wih.T   (K = Kx) ----
  {
    const float* Arow = Ain + (size_t)(r0 + lm) * lda + half * 2;
    const float* WR = Wih + (size_t)(c0 + lm) * Kx + half * 2;
    const float* WZ = WR + (size_t)H_DIM * Kx;
    const float* WN = WZ + (size_t)H_DIM * Kx;
    for (int k = 0; k < Kx; k += 4) {
      v2f a  = *(const v2f*)(Arow + k);
      v2f bR = *(const v2f*)(WR + k);
      v2f bZ = *(const v2f*)(WZ + k);
      v2f bN = *(const v2f*)(WN + k);
      accR   = wmma_f32(a, bR, accR);
      accZ   = wmma_f32(a, bZ, accZ);
      accGin = wmma_f32(a, bN, accGin);
    }
  }
  // ---- hidden-gate GEMM: gh = h_{t-1} @ Whh.T   (K = H) ----
  {
    const float* Hrow = Hprev + (size_t)(r0 + lm) * H_DIM + half * 2;
    const float* WR = Whh + (size_t)(c0 + lm) * H_DIM + half * 2;
    const float* WZ = WR + (size_t)H_DIM * H_DIM;
    const float* WN = WZ + (size_t)H_DIM * H_DIM;
    for (int k = 0; k < H_DIM; k += 4) {
      v2f a  = *(const v2f*)(Hrow + k);
      v2f bR = *(const v2f*)(WR + k);
      v2f bZ = *(const v2f*)(WZ + k);
      v2f bN = *(const v2f*)(WN + k);
      accR   = wmma_f32(a, bR, accR);
      accZ   = wmma_f32(a, bZ, accZ);
      accGhn = wmma_f32(a, bN, accGhn);
    }
  }

  // ---- fused GRU epilogue on the 16x16 tile ----
  const int col = c0 + lm;
  const float br = bih[col]             + bhh[col];
  const float bz = bih[H_DIM + col]     + bhh[H_DIM + col];
  const float bin = bih[2 * H_DIM + col];
  const float bhn = bhh[2 * H_DIM + col];
#pragma unroll
  for (int v = 0; v < 8; ++v) {
    const int row = r0 + v + 8 * half;
    const float r = sigmoid_f(accR[v] + br);
    const float z = sigmoid_f(accZ[v] + bz);
    const float n = tanhf(accGin[v] + bin + r * (accGhn[v] + bhn));
    const float hp = Hprev[(size_t)row * H_DIM + col];
    const float h  = (1.0f - z) * n + z * hp;
    Hnew[(size_t)row * H_DIM + col] = h;
    if (seqOut) seqOut[(size_t)row * seqStride + col] = h;
  }
}

// ---------------------------------------------------------------------------
// BN statistics: deterministic two-level chunked column reduction over seq.
// ---------------------------------------------------------------------------
__global__ void bn_partial_kernel(const float* __restrict__ seq,
                                  float* __restrict__ psum,
                                  float* __restrict__ psq) {
  const int j = (int)threadIdx.x;       // column 0..H-1
  const int c = (int)blockIdx.x;        // chunk 0..NCHUNK-1
  const int ROWS = (B_DIM * T_DIM) / NCHUNK;   // 512 rows per chunk
  size_t base = (size_t)c * ROWS * H_DIM + j;
  float s = 0.0f, ss = 0.0f;
  for (int r = 0; r < ROWS; ++r) {
    float v = seq[base + (size_t)r * H_DIM];
    s += v; ss += v * v;
  }
  psum[c * H_DIM + j] = s;
  psq[c * H_DIM + j]  = ss;
}

__global__ void bn_stats_kernel(const float* __restrict__ psum,
                                const float* __restrict__ psq,
                                const float* __restrict__ gamma,
                                const float* __restrict__ beta,
                                float* __restrict__ scale,
                                float* __restrict__ shift) {
  const int j = (int)threadIdx.x;
  float s = 0.0f, ss = 0.0f;
  for (int c = 0; c < NCHUNK; ++c) {
    s += psum[c * H_DIM + j];
    ss += psq[c * H_DIM + j];
  }
  const float inv = 1.0f / (float)(B_DIM * T_DIM);
  const float mean = s * inv;
  const float var  = ss * inv - mean * mean;   // biased variance
  const float rstd = rsqrtf(var + EPS_BN);
  const float sc = rstd * gamma[j];
  scale[j] = sc;
  shift[j] = beta[j] - mean * sc;
}

// normalize + hardtanh(-2,2) + temporal mean  ->  summary (B,H)
__global__ void summary_kernel(const float* __restrict__ seq,
                               const float* __restrict__ scale,
                               const float* __restrict__ shift,
                               float* __restrict__ summary) {
  const int j = (int)threadIdx.x;
  const int b = (int)blockIdx.x;
  const float sc = scale[j], sh = shift[j];
  size_t base = (size_t)b * T_DIM * H_DIM + j;
  float acc = 0.0f;
  for (int t = 0; t < T_DIM; ++t) {
    float v = seq[base + (size_t)t * H_DIM] * sc + sh;
    v = fminf(2.0f, fmaxf(-2.0f, v));
    acc += v;
  }
  summary[b * H_DIM + j] = acc * (1.0f / (float)T_DIM);
}

// out (B,OUT) = summary (B,H) @ fc_w.T (H,OUT) + fc_b  — fp32 WMMA
__global__ __launch_bounds__(32) void fc_kernel(const float* __restrict__ Sm,
                                                const float* __restrict__ W,
                                                const float* __restrict__ bias,
                                                float* __restrict__ out) {
  const int lane = (int)threadIdx.x;
  const int lm   = lane & 15;
  const int half = lane >> 4;
  const int r0   = (int)blockIdx.x << 4;
  const int c0   = (int)blockIdx.y << 4;
  v8f acc = {};
  const float* Arow = Sm + (size_t)(r0 + lm) * H_DIM + half * 2;
  const float* Wrow = W + (size_t)(c0 + lm) * H_DIM + half * 2;
  for (int k = 0; k < H_DIM; k += 4) {
    v2f a = *(const v2f*)(Arow + k);
    v2f b = *(const v2f*)(Wrow + k);
    acc = wmma_f32(a, b, acc);
  }
  const float bv = bias[c0 + lm];
#pragma unroll
  for (int v = 0; v < 8; ++v) {
    const int row = r0 + v + 8 * half;
    out[(size_t)row * OUT_DIM + (c0 + lm)] = acc[v] + bv;
  }
}

__global__ void zero_kernel(float* __restrict__ p, int n) {
  int i = (int)(blockIdx.x * blockDim.x + threadIdx.x);
  if (i < n) p[i] = 0.0f;
}

// ---------------------------------------------------------------------------
extern "C" void kernel_launch(void* const* d_in, const int* in_sizes, int n_in,
                              void* d_out, int out_size, void* d_ws, size_t ws_size,
                              hipStream_t stream) {
  (void)in_sizes; (void)n_in; (void)out_size; (void)ws_size;
  const float* x     = (const float*)d_in[0];
  const float* w_ih1 = (const float*)d_in[1];
  const float* w_hh1 = (const float*)d_in[2];
  const float* b_ih1 = (const float*)d_in[3];
  const float* b_hh1 = (const float*)d_in[4];
  const float* w_ih2 = (const float*)d_in[5];
  const float* w_hh2 = (const float*)d_in[6];
  const float* b_ih2 = (const float*)d_in[7];
  const float* b_hh2 = (const float*)d_in[8];
  const float* gamma = (const float*)d_in[9];
  const float* beta  = (const float*)d_in[10];
  const float* fc_w  = (const float*)d_in[11];
  const float* fc_b  = (const float*)d_in[12];
  float* out = (float*)d_out;

  const size_t BH  = (size_t)B_DIM * H_DIM;           // 65536
  const size_t SEQ = (size_t)B_DIM * T_DIM * H_DIM;   // 33.5M floats
  float* ws      = (float*)d_ws;
  float* seq     = ws;                 // (B,T,H)
  float* h1      = seq + SEQ;          // 2 x (B,H) double buffer
  float* h2      = h1 + 2 * BH;        // 2 x (B,H) double buffer
  float* psum    = h2 + 2 * BH;        // NCHUNK x H
  float* psq     = psum + (size_t)NCHUNK * H_DIM;
  float* scale   = psq + (size_t)NCHUNK * H_DIM;
  float* shift   = scale + H_DIM;
  float* summary = shift + H_DIM;      // (B,H)

  // zero both hidden-state double buffers (t=0 reads buffer 0)
  {
    int n = (int)(4 * BH);
    zero_kernel<<<(n + 255) / 256, 256, 0, stream>>>(h1, n);
  }

  const dim3 stepGrid(B_DIM / 16, H_DIM / 16);  // 16 x 16 waves
  for (int t = 0; t < T_DIM; ++t) {
    float* h1r = h1 + (size_t)(t & 1) * BH;
    float* h1w = h1 + (size_t)((t + 1) & 1) * BH;
    float* h2r = h2 + (size_t)(t & 1) * BH;
    float* h2w = h2 + (size_t)((t + 1) & 1) * BH;
    // layer 1: x_t rows live at stride T*IN, base offset t*IN
    gru_step_kernel<<<stepGrid, 32, 0, stream>>>(
        x + (size_t)t * IN_DIM, T_DIM * IN_DIM, IN_DIM,
        h1r, w_ih1, w_hh1, b_ih1, b_hh1, h1w, nullptr, 0);
    // layer 2: consumes freshly written h1, emits seq_hidden[:, t, :]
    gru_step_kernel<<<stepGrid, 32, 0, stream>>>(
        h1w, H_DIM, H_DIM,
        h2r, w_ih2, w_hh2, b_ih2, b_hh2, h2w,
        seq + (size_t)t * H_DIM, T_DIM * H_DIM);
  }

  bn_partial_kernel<<<NCHUNK, H_DIM, 0, stream>>>(seq, psum, psq);
  bn_stats_kernel<<<1, H_DIM, 0, stream>>>(psum, psq, gamma, beta, scale, shift);
  summary_kernel<<<B_DIM, H_DIM, 0, stream>>>(seq, scale, shift, summary);
  fc_kernel<<<dim3(B_DIM / 16, OUT_DIM / 16), 32, 0, stream>>>(summary, fc_w, fc_b, out);
}